// SphericalChebBN2_6786048327768
// MI455X (gfx1250) — compile-verified
//
#include <hip/hip_runtime.h>
#include <hip/hip_bf16.h>

typedef float v2f __attribute__((ext_vector_type(2)));
typedef float v8f __attribute__((ext_vector_type(8)));

#define N_VERT 196608
#define N_EDGE 1572864
#define BATCH  4
#define M_ROWS ((long)BATCH * N_VERT)   // 786432
#define COUT   64

// ---------------------------------------------------------------------------
// Build row_ptr (CSR offsets) from sorted row[] — deterministic, no atomics.
// ---------------------------------------------------------------------------
__global__ void build_row_ptr(const int* __restrict__ row, int* __restrict__ row_ptr,
                              int E, int N) {
    int e = blockIdx.x * blockDim.x + threadIdx.x;
    if (e > E) return;
    if (e == E) {
        int last = row[E - 1];
        for (int r = last + 1; r <= N; ++r) row_ptr[r] = E;
        return;
    }
    int rc = row[e];
    int rp = (e == 0) ? -1 : row[e - 1];
    for (int r = rp + 1; r <= rc; ++r) row_ptr[r] = e;
}

// ---------------------------------------------------------------------------
// Fused SPMM + Chebyshev combine: out = alpha * (L @ src) + beta * prev2
// One wave per (vertex row, batch). CPL = channels per lane (C = 32*CPL).
// `out` may alias `prev2` (element read before write by the same lane).
// ---------------------------------------------------------------------------
template <int CPL>
__global__ __launch_bounds__(256) void spmm_cheb(
    const float* __restrict__ src, const float* __restrict__ prev2,
    float* __restrict__ out, const int* __restrict__ row_ptr,
    const int* __restrict__ col, const float* __restrict__ w,
    float alpha, float beta) {
    const int C    = CPL * 32;
    const int lane = threadIdx.x & 31;
    const int wave = threadIdx.x >> 5;
    const int p    = blockIdx.x * 8 + wave;   // p = r*BATCH + b
    const int r    = p >> 2;                  // BATCH == 4
    const int b    = p & 3;

    const int e0 = row_ptr[r];
    const int e1 = row_ptr[r + 1];
    const size_t base_b = (size_t)b * N_VERT * C;

    float acc0 = 0.f, acc1 = 0.f;
    for (int es = e0; es < e1; es += 32) {
        int n = e1 - es; if (n > 32) n = 32;
        int   cl = 0; float wl = 0.f;
        if (lane < n) { cl = col[es + lane]; wl = w[es + lane]; }
        for (int j = 0; j < n; ++j) {
            int   cj = __shfl(cl, j, 32);
            float wj = __shfl(wl, j, 32);
            const float* xp = src + base_b + (size_t)cj * C;
            acc0 += wj * xp[lane];
            if (CPL == 2) acc1 += wj * xp[lane + 32];
        }
    }
    const size_t o = base_b + (size_t)r * C;
    float r0 = alpha * acc0, r1 = alpha * acc1;
    if (prev2 != nullptr) {
        r0 += beta * prev2[o + lane];
        if (CPL == 2) r1 += beta * prev2[o + lane + 32];
    }
    out[o + lane] = r0;
    if (CPL == 2) out[o + lane + 32] = r1;
}

// ---------------------------------------------------------------------------
// WMMA GEMM-accumulate: acc[M,64] (+)= T[M,CIN] @ theta[CIN,64]
//
// theta is staged once per workgroup into LDS in *fragment order*: for each
// k-step and lane, the 4 column-tile B fragments (v2f each) are contiguous
// (32 bytes), so the inner loop reads B with 2x ds_load_b128 instead of
// 8x global_load_b32.  Each wave owns 2 consecutive 16-row M tiles and reuses
// the B fragments across both A fragments -> 8 independent WMMA per k-step.
// f32 fragment layouts per CDNA5 ISA 7.12.2 (A 16x4, B 4x16, C/D 16x16).
// ---------------------------------------------------------------------------
template <int CIN>
__global__ __launch_bounds__(256) void cheb_gemm_acc(
    const float* __restrict__ T, const float* __restrict__ theta,
    float* __restrict__ acc, int accumulate) {
    constexpr int KSTEPS = CIN / 4;
    __shared__ v2f ldsB[KSTEPS * 128];   // [kstep][lane][ftile] -> v2f

    // ---- stage theta into LDS in fragment order ----
    for (int idx = threadIdx.x; idx < KSTEPS * 128; idx += 256) {
        const int f    = idx & 3;
        const int ln   = (idx >> 2) & 31;
        const int kk   = idx >> 7;
        const int kgS  = (ln >> 4) * 2;      // K sub-group of that lane
        const int ncS  = ln & 15;
        const int r0   = kk * 4 + kgS;       // theta row of frag .x
        v2f b;
        b.x = theta[(size_t)r0 * COUT + f * 16 + ncS];
        b.y = theta[(size_t)(r0 + 1) * COUT + f * 16 + ncS];
        ldsB[idx] = b;
    }
    __syncthreads();

    const int lane = threadIdx.x & 31;
    const int wave = threadIdx.x >> 5;
    const long m0  = ((long)blockIdx.x * 8 + wave) * 32;   // 2 tiles of 16 rows
    const long m1  = m0 + 16;

    const int mr  = lane & 15;        // A row within tile
    const int kg  = (lane >> 4) * 2;  // K sub-group for A frags
    const int hi8 = (lane >> 4) * 8;  // C/D row offset
    const int nc  = lane & 15;        // C/D column within 16-tile

    v8f c00 = {}, c01 = {}, c02 = {}, c03 = {};
    v8f c10 = {}, c11 = {}, c12 = {}, c13 = {};
    if (accumulate) {
#pragma unroll
        for (int r = 0; r < 8; ++r) {
            const float* p0 = acc + (m0 + hi8 + r) * COUT + nc;
            const float* p1 = acc + (m1 + hi8 + r) * COUT + nc;
            c00[r] = p0[0]; c01[r] = p0[16]; c02[r] = p0[32]; c03[r] = p0[48];
            c10[r] = p1[0]; c11[r] = p1[16]; c12[r] = p1[32]; c13[r] = p1[48];
        }
    }

    const v2f* lb = ldsB + lane * 4;
#pragma unroll
    for (int kk = 0; kk < KSTEPS; ++kk) {
        v2f a0 = *(const v2f*)(T + (m0 + mr) * CIN + kk * 4 + kg);
        v2f a1 = *(const v2f*)(T + (m1 + mr) * CIN + kk * 4 + kg);
        v2f b0 = lb[kk * 128 + 0];
        v2f b1 = lb[kk * 128 + 1];
        v2f b2 = lb[kk * 128 + 2];
        v2f b3 = lb[kk * 128 + 3];
        c00 = __builtin_amdgcn_wmma_f32_16x16x4_f32(false, a0, false, b0, (short)0, c00, false, false);
        c10 = __builtin_amdgcn_wmma_f32_16x16x4_f32(false, a1, false, b0, (short)0, c10, false, false);
        c01 = __builtin_amdgcn_wmma_f32_16x16x4_f32(false, a0, false, b1, (short)0, c01, false, false);
        c11 = __builtin_amdgcn_wmma_f32_16x16x4_f32(false, a1, false, b1, (short)0, c11, false, false);
        c02 = __builtin_amdgcn_wmma_f32_16x16x4_f32(false, a0, false, b2, (short)0, c02, false, false);
        c12 = __builtin_amdgcn_wmma_f32_16x16x4_f32(false, a1, false, b2, (short)0, c12, false, false);
        c03 = __builtin_amdgcn_wmma_f32_16x16x4_f32(false, a0, false, b3, (short)0, c03, false, false);
        c13 = __builtin_amdgcn_wmma_f32_16x16x4_f32(false, a1, false, b3, (short)0, c13, false, false);
    }

#pragma unroll
    for (int r = 0; r < 8; ++r) {
        float* p0 = acc + (m0 + hi8 + r) * COUT + nc;
        float* p1 = acc + (m1 + hi8 + r) * COUT + nc;
        p0[0] = c00[r]; p0[16] = c01[r]; p0[32] = c02[r]; p0[48] = c03[r];
        p1[0] = c10[r]; p1[16] = c11[r]; p1[32] = c12[r]; p1[48] = c13[r];
    }
}

// ---------------------------------------------------------------------------
// BatchNorm stats: deterministic two-stage reduction over M rows, 64 channels.
// ---------------------------------------------------------------------------
__global__ __launch_bounds__(256) void bn_partial(const float* __restrict__ x,
                                                  float* __restrict__ psum,
                                                  float* __restrict__ psq, long M) {
    __shared__ float ss[256], sq[256];
    const int c  = threadIdx.x & 63;
    const int rl = threadIdx.x >> 6;  // 0..3
    float s = 0.f, q = 0.f;
    for (long r = (long)blockIdx.x * 4 + rl; r < M; r += (long)gridDim.x * 4) {
        float v = x[r * 64 + c];
        s += v; q += v * v;
    }
    ss[threadIdx.x] = s; sq[threadIdx.x] = q;
    __syncthreads();
    if (rl == 0) {
        for (int j = 1; j < 4; ++j) { s += ss[c + 64 * j]; q += sq[c + 64 * j]; }
        psum[blockIdx.x * 64 + c] = s;
        psq [blockIdx.x * 64 + c] = q;
    }
}

__global__ __launch_bounds__(256) void bn_finalize(const float* __restrict__ psum,
                                                   const float* __restrict__ psq,
                                                   float* __restrict__ mean,
                                                   float* __restrict__ rstd,
                                                   int P, long M) {
    __shared__ float ss[256], sq[256];
    const int c = threadIdx.x & 63;
    const int j = threadIdx.x >> 6;
    float s = 0.f, q = 0.f;
    for (int p = j; p < P; p += 4) { s += psum[p * 64 + c]; q += psq[p * 64 + c]; }
    ss[threadIdx.x] = s; sq[threadIdx.x] = q;
    __syncthreads();
    if (j == 0) {
        for (int t = 1; t < 4; ++t) { s += ss[c + 64 * t]; q += sq[c + 64 * t]; }
        float m = s / (float)M;
        float v = q / (float)M - m * m;   // biased variance (jnp.var default)
        mean[c] = m;
        rstd[c] = rsqrtf(v + 1e-5f);
    }
}

__global__ void bn_relu_apply(float* __restrict__ x, const float* __restrict__ mean,
                              const float* __restrict__ rstd,
                              const float* __restrict__ gamma,
                              const float* __restrict__ beta, long total) {
    long i = (long)blockIdx.x * blockDim.x + threadIdx.x;
    const long stride = (long)gridDim.x * blockDim.x;
    for (; i < total; i += stride) {
        int c = (int)(i & 63);
        float v = (x[i] - mean[c]) * rstd[c] * gamma[c] + beta[c];
        x[i] = v > 0.f ? v : 0.f;
    }
}

// ---------------------------------------------------------------------------
extern "C" void kernel_launch(void* const* d_in, const int* in_sizes, int n_in,
                              void* d_out, int out_size, void* d_ws, size_t ws_size,
                              hipStream_t stream) {
    const float* x      = (const float*)d_in[0];   // [4, N, 32]
    const int*   row    = (const int*)  d_in[1];   // [E] sorted
    const int*   col    = (const int*)  d_in[2];   // [E]
    const float* ew     = (const float*)d_in[3];   // [E]
    const float* theta1 = (const float*)d_in[4];   // [5, 32, 64]
    const float* gamma1 = (const float*)d_in[5];
    const float* beta1  = (const float*)d_in[6];
    const float* theta2 = (const float*)d_in[7];   // [5, 64, 64]
    const float* gamma2 = (const float*)d_in[8];
    const float* beta2  = (const float*)d_in[9];
    float* out = (float*)d_out;                    // [4, N, 64]

    const long M = M_ROWS;
    const size_t bufBytes = (size_t)M * 64 * sizeof(float);   // 201.3 MB

    char* ws = (char*)d_ws;
    float* bufA   = (float*)ws;              ws += bufBytes;      // Cheb term scratch
    float* bufH   = (float*)ws;              ws += bufBytes;      // layer-1 activation
    int*   rowptr = (int*)ws;                ws += (size_t)(N_VERT + 1) * sizeof(int);
    float* psum   = (float*)ws;              ws += 256 * 64 * sizeof(float);
    float* psq    = (float*)ws;              ws += 256 * 64 * sizeof(float);
    float* meanb  = (float*)ws;              ws += 64 * sizeof(float);
    float* rstdb  = (float*)ws;              ws += 64 * sizeof(float);

    const int spmmBlocks = (int)(M_ROWS / 8);    // 1 wave per (row,batch)
    const int gemmBlocks = (int)(M / 256);       // 8 waves x 2 tiles x 16 rows

    build_row_ptr<<<(N_EDGE + 1 + 255) / 256, 256, 0, stream>>>(row, rowptr, N_EDGE, N_VERT);

    // ------------------- Layer 1 (Cin = 32) -------------------
    // acc = bufH ; Cheb buffers rotate between bufA and d_out (scratch for now)
    cheb_gemm_acc<32><<<gemmBlocks, 256, 0, stream>>>(x, theta1 + 0 * 32 * 64, bufH, 0);
    spmm_cheb<1><<<spmmBlocks, 256, 0, stream>>>(x, nullptr, bufA, rowptr, col, ew, 1.f, 0.f);     // T1
    cheb_gemm_acc<32><<<gemmBlocks, 256, 0, stream>>>(bufA, theta1 + 1 * 32 * 64, bufH, 1);
    spmm_cheb<1><<<spmmBlocks, 256, 0, stream>>>(bufA, x, out, rowptr, col, ew, 2.f, -1.f);        // T2
    cheb_gemm_acc<32><<<gemmBlocks, 256, 0, stream>>>(out, theta1 + 2 * 32 * 64, bufH, 1);
    spmm_cheb<1><<<spmmBlocks, 256, 0, stream>>>(out, bufA, bufA, rowptr, col, ew, 2.f, -1.f);     // T3
    cheb_gemm_acc<32><<<gemmBlocks, 256, 0, stream>>>(bufA, theta1 + 3 * 32 * 64, bufH, 1);
    spmm_cheb<1><<<spmmBlocks, 256, 0, stream>>>(bufA, out, out, rowptr, col, ew, 2.f, -1.f);      // T4
    cheb_gemm_acc<32><<<gemmBlocks, 256, 0, stream>>>(out, theta1 + 4 * 32 * 64, bufH, 1);

    bn_partial <<<256, 256, 0, stream>>>(bufH, psum, psq, M);
    bn_finalize<<<1,   256, 0, stream>>>(psum, psq, meanb, rstdb, 256, M);
    bn_relu_apply<<<4096, 256, 0, stream>>>(bufH, meanb, rstdb, gamma1, beta1, M * 64);

    // ------------------- Layer 2 (Cin = 64) -------------------
    // acc = d_out ; Cheb buffers rotate between bufA and bufH (T0 = bufH)
    cheb_gemm_acc<64><<<gemmBlocks, 256, 0, stream>>>(bufH, theta2 + 0 * 64 * 64, out, 0);
    spmm_cheb<2><<<spmmBlocks, 256, 0, stream>>>(bufH, nullptr, bufA, rowptr, col, ew, 1.f, 0.f);  // T1
    cheb_gemm_acc<64><<<gemmBlocks, 256, 0, stream>>>(bufA, theta2 + 1 * 64 * 64, out, 1);
    spmm_cheb<2><<<spmmBlocks, 256, 0, stream>>>(bufA, bufH, bufH, rowptr, col, ew, 2.f, -1.f);    // T2
    cheb_gemm_acc<64><<<gemmBlocks, 256, 0, stream>>>(bufH, theta2 + 2 * 64 * 64, out, 1);
    spmm_cheb<2><<<spmmBlocks, 256, 0, stream>>>(bufH, bufA, bufA, rowptr, col, ew, 2.f, -1.f);    // T3
    cheb_gemm_acc<64><<<gemmBlocks, 256, 0, stream>>>(bufA, theta2 + 3 * 64 * 64, out, 1);
    spmm_cheb<2><<<spmmBlocks, 256, 0, stream>>>(bufA, bufH, bufH, rowptr, col, ew, 2.f, -1.f);    // T4
    cheb_gemm_acc<64><<<gemmBlocks, 256, 0, stream>>>(bufH, theta2 + 4 * 64 * 64, out, 1);

    bn_partial <<<256, 256, 0, stream>>>(out, psum, psq, M);
    bn_finalize<<<1,   256, 0, stream>>>(psum, psq, meanb, rstdb, 256, M);
    bn_relu_apply<<<4096, 256, 0, stream>>>(out, meanb, rstdb, gamma2, beta2, M * 64);
}